// GradientLayer_10900626997309
// MI455X (gfx1250) — compile-verified
//
#include <hip/hip_runtime.h>

typedef float v2f __attribute__((ext_vector_type(2)));
typedef float v8f __attribute__((ext_vector_type(8)));

#define WAVES  8
#define STG    132          /* 128 + 4 floats pad -> conflict-free row access */
#define MASS_C 2.0f

__launch_bounds__(256, 1)
__global__ void pinn_mlp_wmma_f32(
    const float* __restrict__ x,
    const float* __restrict__ W0, const float* __restrict__ b0,
    const float* __restrict__ W1, const float* __restrict__ b1,
    const float* __restrict__ W2, const float* __restrict__ b2,
    const float* __restrict__ W3, const float* __restrict__ b3,
    float* __restrict__ out, int B)
{
    // Pre-swizzled B operands for v_wmma_f32_16x16x4_f32, whole net resident in LDS.
    __shared__ float w1p[16384];      // 64 KB: W1 packed (8 n-tiles x 32 k-slices)
    __shared__ float w2p[16384];      // 64 KB: W2 packed
    __shared__ float w3p[2048];       //  8 KB: W3 packed, zero-padded to 16 cols
    __shared__ float w0s[512];        // W0 raw [4][128]
    __shared__ float b0s[128], b1s[128], b2s[128], b3s[8];
    __shared__ float stg[WAVES][16 * STG];  // per-wave 16x128 C-tile staging
    __shared__ float ysd[WAVES][256];       // per-wave tanh / (1-y^2) caches

    const int tid  = threadIdx.x;
    const int lane = tid & 31;
    const int wave = tid >> 5;

    // ---- pack weights into WMMA B-operand order ----
    // slot(n,s,l): lane l, k-slice s, 16-col tile n holds
    //   W[4s+koff+0][16n+(l&15)], W[4s+koff+1][16n+(l&15)], koff=(l<16)?0:2
    for (int slot = tid; slot < 8192; slot += 256) {
        const int l = slot & 31, s = (slot >> 5) & 31, n = slot >> 10;
        const int koff = (l < 16) ? 0 : 2;
        const int col  = (n << 4) | (l & 15);
        const int k0   = 4 * s + koff;
        w1p[slot * 2]     = W1[k0 * 128 + col];
        w1p[slot * 2 + 1] = W1[(k0 + 1) * 128 + col];
        w2p[slot * 2]     = W2[k0 * 128 + col];
        w2p[slot * 2 + 1] = W2[(k0 + 1) * 128 + col];
    }
    for (int slot = tid; slot < 1024; slot += 256) {
        const int l = slot & 31, s = slot >> 5;
        const int koff = (l < 16) ? 0 : 2;
        const int col  = l & 15;
        const int k0   = 4 * s + koff;
        w3p[slot * 2]     = (col < 5) ? W3[k0 * 5 + col]       : 0.0f;
        w3p[slot * 2 + 1] = (col < 5) ? W3[(k0 + 1) * 5 + col] : 0.0f;
    }
    for (int i = tid; i < 512; i += 256) w0s[i] = W0[i];
    if (tid < 128) { b0s[tid] = b0[tid]; b1s[tid] = b1[tid]; b2s[tid] = b2[tid]; }
    if (tid < 8)   b3s[tid] = (tid < 5) ? b3[tid] : 0.0f;
    __syncthreads();

    float* const SW = stg[wave];
    float* const Y  = ysd[wave];
    float* const S  = Y + 128;

    // per-lane channel-row constants (A-operand: lane holds row m = lane&15)
    const int   m      = lane & 15;
    const int   koff   = (lane < 16) ? 0 : 2;
    const bool  primal = (m == 0);
    const bool  fderiv = (m >= 1 && m <= 4);
    const float rowScl = (m == 15) ? 0.0f : 1.0f;  // row 15 = zero pad
    float coef = 0.0f;
    int jr = 0, kr = 0;
    if (m >= 5 && m <= 14) {            // unique Hessian pairs (j<=k)
        coef = 2.0f;
        const int p = m - 5;            // 00,01,02,03,11,12,13,22,23,33
        if (p < 4)      { jr = 0; kr = p;     }
        else if (p < 7) { jr = 1; kr = p - 3; }
        else if (p < 9) { jr = 2; kr = p - 5; }
        else            { jr = 3; kr = 3;     }
    }
    const int mrow0 = fderiv ? (m - 1) : 0;

    v2f A[32];  // A-operand: 16 rows x 128 K, this lane's 64 K-values

    // one 16x16 output tile = one sample; two tiles in flight for XDL ILP
    auto gemm8 = [&](const float* pw) {
        #pragma unroll
        for (int n = 0; n < 8; n += 2) {
            v8f acc0 = {0,0,0,0,0,0,0,0};
            v8f acc1 = {0,0,0,0,0,0,0,0};
            const v2f* bp0 = reinterpret_cast<const v2f*>(pw) + (n * 1024 + lane);
            const v2f* bp1 = bp0 + 1024;
            #pragma unroll
            for (int s = 0; s < 32; ++s) {
                const v2f bv0 = bp0[s * 32];
                const v2f bv1 = bp1[s * 32];
                acc0 = __builtin_amdgcn_wmma_f32_16x16x4_f32(
                           false, A[s], false, bv0, (short)0, acc0, false, false);
                acc1 = __builtin_amdgcn_wmma_f32_16x16x4_f32(
                           false, A[s], false, bv1, (short)0, acc1, false, false);
            }
            const int rb = (lane >= 16) ? 8 : 0;   // C layout: lanes16-31 -> rows r+8
            #pragma unroll
            for (int r = 0; r < 8; ++r) {
                SW[(rb + r) * STG + ((n << 4) | m)]       = acc0[r];
                SW[(rb + r) * STG + (((n + 1) << 4) | m)] = acc1[r];
            }
        }
        __builtin_amdgcn_wave_barrier();
    };

    auto gemm1 = [&](const float* pw) {            // final layer, one padded tile
        v8f acc = {0,0,0,0,0,0,0,0};
        const v2f* bp = reinterpret_cast<const v2f*>(pw) + lane;
        #pragma unroll
        for (int s = 0; s < 32; ++s) {
            const v2f bv = bp[s * 32];
            acc = __builtin_amdgcn_wmma_f32_16x16x4_f32(
                      false, A[s], false, bv, (short)0, acc, false, false);
        }
        const int rb = (lane >= 16) ? 8 : 0;
        #pragma unroll
        for (int r = 0; r < 8; ++r) SW[(rb + r) * STG + m] = acc[r];
        __builtin_amdgcn_wave_barrier();
    };

    // tanh + forward-mode derivative propagation, then rebuild A-operand regs
    auto nonlin = [&](const float* bias) {
        #pragma unroll
        for (int i = 0; i < 4; ++i) {               // cooperative: 4 tanh / lane
            const int c = lane + 32 * i;
            const float y = tanhf(SW[c] + bias[c]); // bias only on primal row
            Y[c] = y;
            S[c] = 1.0f - y * y;
        }
        __builtin_amdgcn_wave_barrier();
        #pragma unroll
        for (int s = 0; s < 32; ++s) {
            #pragma unroll
            for (int t = 0; t < 2; ++t) {
                const int K  = 4 * s + koff + t;
                const float y = Y[K], sv = S[K];
                const float um = SW[m * STG + K];
                const float uj = SW[(1 + jr) * STG + K];
                const float uk = SW[(1 + kr) * STG + K];
                // m=0: y | m=1..4: s*u' | m=5..14: s*(u'' - 2y u'_j u'_k) | m=15: 0
                A[s][t] = primal ? y : rowScl * sv * (um - coef * y * uj * uk);
            }
        }
        __builtin_amdgcn_wave_barrier();
    };

    const int gwave   = blockIdx.x * WAVES + wave;
    const int gstride = gridDim.x * WAVES;

    for (int smp = gwave; smp < B; smp += gstride) {
        // ===== layer 0: u0 = x@W0+b0; tangents = rows of W0; curvature = 0 =====
        const float4 xv = reinterpret_cast<const float4*>(x)[smp];
        #pragma unroll
        for (int i = 0; i < 4; ++i) {
            const int c = lane + 32 * i;
            const float u = xv.x * w0s[c] + xv.y * w0s[128 + c] +
                            xv.z * w0s[256 + c] + xv.w * w0s[384 + c] + b0s[c];
            const float y = tanhf(u);
            Y[c] = y;
            S[c] = 1.0f - y * y;
        }
        __builtin_amdgcn_wave_barrier();
        #pragma unroll
        for (int s = 0; s < 32; ++s) {
            #pragma unroll
            for (int t = 0; t < 2; ++t) {
                const int K = 4 * s + koff + t;
                const float y = Y[K], sv = S[K];
                float f;
                if (primal)      f = y;
                else if (fderiv) f = sv * w0s[mrow0 * 128 + K];
                else             f = rowScl * (-2.0f) * y * sv *
                                     w0s[jr * 128 + K] * w0s[kr * 128 + K];
                A[s][t] = f;
            }
        }
        __builtin_amdgcn_wave_barrier();

        gemm8(w1p);  nonlin(b1s);     // h1 = tanh(.@W1+b1) with derivatives
        gemm8(w2p);  nonlin(b2s);     // h2 = tanh(.@W2+b2) with derivatives
        gemm1(w3p);                   // o rows: 0=primal, 1..4=d/dz_k, 5..14=Hess

        // ===== epilogue: assemble the reference's 38 per-sample outputs =====
        if (lane == 0) {
            auto O = [&](int r, int c) { return SW[r * STG + c]; };
            const float n_  = O(0,0) + b3s[0];
            const float vv0 = O(0,1) + b3s[1];
            const float vv1 = O(0,2) + b3s[2];
            const float vv2 = O(0,3) + b3s[3];
            const float Fi_ = O(0,4) + b3s[4];
            const float nt  = O(1,0);
            const float ng0 = O(2,0), ng1 = O(3,0), ng2 = O(4,0);
            const float vv[3] = {vv0, vv1, vv2};
            float jj[3], jdv[3];
            #pragma unroll
            for (int i = 0; i < 3; ++i) {
                jj[i] = n_ * vv[i];
                float sum = 0.0f;
                #pragma unroll
                for (int d = 1; d <= 3; ++d)
                    sum += O(1 + d, 0) * vv[i] + n_ * O(1 + d, 1 + i);
                jdv[i] = sum;
            }
            // Hessian rows: 5:(00) 6:(01) 7:(02) 8:(03) 9:(11) 10:(12) 11:(13) 12:(22) 13:(23) 14:(33)
            const float Fg0 = O(2,4), Fg1 = O(3,4), Fg2 = O(4,4);
            const float Fl0 = O(6,4)  + O(7,4)  + O(8,4);    // sum_b HFi[a][1..3]
            const float Fl1 = O(9,4)  + O(10,4) + O(11,4);
            const float Fl2 = O(10,4) + O(12,4) + O(13,4);
            const float Fl3 = O(11,4) + O(13,4) + O(14,4);
            const float vlap = O(6,0) + O(10,0) + O(13,0);   // Hn[0,1]+Hn[1,2]+Hn[2,3]
            const float vd0  = O(9,0)  + O(10,0) + O(11,0);
            const float vd1  = O(10,0) + O(12,0) + O(13,0);
            const float vd2  = O(11,0) + O(13,0) + O(14,0);

            float* op = out + smp;
            op[0*B]  = n_;            op[1*B]  = nt;
            op[2*B]  = ng0;           op[3*B]  = ng1;           op[4*B]  = ng2;
            op[5*B]  = jj[0];         op[6*B]  = jj[1];         op[7*B]  = jj[2];
            op[8*B]  = jdv[0];        op[9*B]  = jdv[1];        op[10*B] = jdv[2];
            op[11*B] = Fi_;
            op[12*B] = Fg0;           op[13*B] = Fg1;           op[14*B] = Fg2;
            op[15*B] = Fl0;           op[16*B] = Fl1;           op[17*B] = Fl2;  op[18*B] = Fl3;
            op[19*B] = vv0;           op[20*B] = vv1;           op[21*B] = vv2;
            op[22*B] = nt;                                   // v_t  (v_jac = n_jac)
            op[23*B] = vv0 * ng0;     op[24*B] = vv1 * ng1;     op[25*B] = vv2 * ng2;
            op[26*B] = vlap;
            op[27*B] = vd0;           op[28*B] = vd1;           op[29*B] = vd2;
            op[30*B] = MASS_C * n_;   op[31*B] = MASS_C * nt;
            op[32*B] = MASS_C * ng0;  op[33*B] = MASS_C * ng1;  op[34*B] = MASS_C * ng2;
            op[35*B] = MASS_C * jdv[0]; op[36*B] = MASS_C * jdv[1]; op[37*B] = MASS_C * jdv[2];
        }
    }
}

extern "C" void kernel_launch(void* const* d_in, const int* in_sizes, int n_in,
                              void* d_out, int out_size, void* d_ws, size_t ws_size,
                              hipStream_t stream) {
    const float* x  = (const float*)d_in[0];
    const float* W0 = (const float*)d_in[1];
    const float* b0 = (const float*)d_in[2];
    const float* W1 = (const float*)d_in[3];
    const float* b1 = (const float*)d_in[4];
    const float* W2 = (const float*)d_in[5];
    const float* b2 = (const float*)d_in[6];
    const float* W3 = (const float*)d_in[7];
    const float* b3 = (const float*)d_in[8];
    const int B = in_sizes[0] / 4;
    (void)n_in; (void)out_size; (void)d_ws; (void)ws_size;
    hipLaunchKernelGGL(pinn_mlp_wmma_f32, dim3(256), dim3(256), 0, stream,
                       x, W0, b0, W1, b1, W2, b2, W3, b3, (float*)d_out, B);
}